// MLPAttention_18494129177217
// MI455X (gfx1250) — compile-verified
//
#include <hip/hip_runtime.h>
#include <math.h>

typedef float v2f __attribute__((ext_vector_type(2)));
typedef float v8f __attribute__((ext_vector_type(8)));

#define T_DIM 128
#define B_DIM 8
#define S_DIM 128
#define H_DIM 512
#define A_DIM 512
#define WAVES_PER_BLOCK 4
#define HCHUNK 64

// ---------------------------------------------------------------------------
// Generic fp32 WMMA GEMM:  C[M x N] = act( Ain[M x K] * B + bias )
//   Ain row r, col k:  k <  Ksplit -> Xa[r*lda_a + k]
//                      k >= Ksplit -> Xb[r*lda_b + (k - Ksplit)]   (HAS_XB)
//   B-matrix:  B_ROWMAJOR ? B[k][n] = W[k*ldw + n]   (row-major K x N)
//                         : B[k][n] = W[n*ldw + k]   (i.e. X * W^T, W is N x K)
//   Batched over blockIdx.y via element strides (Xa, W, C only).
// One wave computes a 32x64 strip: 8 accumulators (2 row-tiles x 4 col-tiles).
// K-loop is unrolled x8 into two static phases (A/B fragment sets) so next
// phase's loads overlap current phase's WMMAs, with a SINGLE-basic-block loop
// body (no in-loop branches -> no accumulator copy storms at CFG merges).
// Requires K % 8 == 0 (holds: 512, 1024, 128).
// ---------------------------------------------------------------------------
template <bool B_ROWMAJOR, bool ACT_TANH, bool HAS_XB, bool HAS_BIAS>
__global__ __launch_bounds__(128) void wmma_gemm_f32(
    const float* __restrict__ Xa, int lda_a,
    const float* __restrict__ Xb, int lda_b, int Ksplit,
    const float* __restrict__ W, int ldw,
    const float* __restrict__ bias,
    float* __restrict__ C, int ldc,
    int M, int N, int K,
    long strideXa, long strideW, long strideC)
{
    const int bb = blockIdx.y;
    Xa += (long)bb * strideXa;
    W  += (long)bb * strideW;
    C  += (long)bb * strideC;

    const int lane = threadIdx.x & 31;
    const int wave = threadIdx.x >> 5;
    const int nStrips = N >> 6;  // 64-column strips
    const int gw = blockIdx.x * WAVES_PER_BLOCK + wave;
    const int rowTile = gw / nStrips;   // 32-row tiles
    const int strip   = gw % nStrips;
    if (rowTile * 32 >= M) return;  // wave-uniform: EXEC stays all-ones for WMMA

    const int hi = lane >> 4;   // 0 or 1
    const int lo = lane & 15;
    const int arow0 = rowTile * 32 + lo;   // A-fragment rows (M index)
    const int arow1 = arow0 + 16;
    const int colBase = strip * 64;

    auto loadA = [&](int row, int kk) -> v2f {
        const float* Xp = Xa; int kloc = kk; int ldx = lda_a;
        if (HAS_XB && kk >= Ksplit) { Xp = Xb; kloc = kk - Ksplit; ldx = lda_b; }
        return *(const v2f*)(Xp + (long)row * ldx + kloc + 2 * hi);
    };
    auto loadBf = [&](int kk, int ci) -> v2f {
        const int kb = kk + 2 * hi;
        const int n = colBase + ci * 16 + lo;
        v2f r;
        if (B_ROWMAJOR) { r.x = W[(long)kb * ldw + n]; r.y = W[(long)(kb + 1) * ldw + n]; }
        else            { r = *(const v2f*)(W + (long)n * ldw + kb); }
        return r;
    };

    // Pre-loop weight-stream prefetches (-> global_prefetch_b8), branch-free.
    {
        const int n0 = colBase + lo;
        if (B_ROWMAJOR) {
            __builtin_prefetch(W + (long)(2 * hi) * ldw + n0, 0, 3);
            __builtin_prefetch(W + (long)(32 + 2 * hi) * ldw + n0, 0, 3);
        } else {
            __builtin_prefetch(W + (long)n0 * ldw, 0, 3);
            __builtin_prefetch(W + (long)(n0 + 32) * ldw, 0, 3);
        }
    }

    v8f c00 = {}, c01 = {}, c02 = {}, c03 = {};
    v8f c10 = {}, c11 = {}, c12 = {}, c13 = {};

    // Phase-A prologue fragments (k = 0)
    v2f a0A = loadA(arow0, 0), a1A = loadA(arow1, 0);
    v2f b0A = loadBf(0, 0), b1A = loadBf(0, 1), b2A = loadBf(0, 2), b3A = loadBf(0, 3);

    for (int kk = 0; kk < K; kk += 8) {
        const int k1 = kk + 4;                          // phase-B k
        const int k2 = (kk + 8 < K) ? (kk + 8) : kk;    // next phase-A k (branchless clamp)

        // issue phase-B loads, then consume phase-A fragments
        v2f a0B = loadA(arow0, k1), a1B = loadA(arow1, k1);
        v2f b0B = loadBf(k1, 0), b1B = loadBf(k1, 1), b2B = loadBf(k1, 2), b3B = loadBf(k1, 3);

        c00 = __builtin_amdgcn_wmma_f32_16x16x4_f32(false, a0A, false, b0A, (short)0, c00, false, false);
        c01 = __builtin_amdgcn_wmma_f32_16x16x4_f32(false, a0A, false, b1A, (short)0, c01, false, false);
        c02 = __builtin_amdgcn_wmma_f32_16x16x4_f32(false, a0A, false, b2A, (short)0, c02, false, false);
        c03 = __builtin_amdgcn_wmma_f32_16x16x4_f32(false, a0A, false, b3A, (short)0, c03, false, false);
        c10 = __builtin_amdgcn_wmma_f32_16x16x4_f32(false, a1A, false, b0A, (short)0, c10, false, false);
        c11 = __builtin_amdgcn_wmma_f32_16x16x4_f32(false, a1A, false, b1A, (short)0, c11, false, false);
        c12 = __builtin_amdgcn_wmma_f32_16x16x4_f32(false, a1A, false, b2A, (short)0, c12, false, false);
        c13 = __builtin_amdgcn_wmma_f32_16x16x4_f32(false, a1A, false, b3A, (short)0, c13, false, false);

        // issue next phase-A loads, then consume phase-B fragments
        a0A = loadA(arow0, k2); a1A = loadA(arow1, k2);
        b0A = loadBf(k2, 0); b1A = loadBf(k2, 1); b2A = loadBf(k2, 2); b3A = loadBf(k2, 3);

        c00 = __builtin_amdgcn_wmma_f32_16x16x4_f32(false, a0B, false, b0B, (short)0, c00, false, false);
        c01 = __builtin_amdgcn_wmma_f32_16x16x4_f32(false, a0B, false, b1B, (short)0, c01, false, false);
        c02 = __builtin_amdgcn_wmma_f32_16x16x4_f32(false, a0B, false, b2B, (short)0, c02, false, false);
        c03 = __builtin_amdgcn_wmma_f32_16x16x4_f32(false, a0B, false, b3B, (short)0, c03, false, false);
        c10 = __builtin_amdgcn_wmma_f32_16x16x4_f32(false, a1B, false, b0B, (short)0, c10, false, false);
        c11 = __builtin_amdgcn_wmma_f32_16x16x4_f32(false, a1B, false, b1B, (short)0, c11, false, false);
        c12 = __builtin_amdgcn_wmma_f32_16x16x4_f32(false, a1B, false, b2B, (short)0, c12, false, false);
        c13 = __builtin_amdgcn_wmma_f32_16x16x4_f32(false, a1B, false, b3B, (short)0, c13, false, false);
    }

    // Store: C/D layout — VGPR vg holds row base + vg + 8*hi, col = lane%16
    v8f accA[4] = {c00, c01, c02, c03};   // rows rowTile*32 + ...
    v8f accB[4] = {c10, c11, c12, c13};   // rows rowTile*32 + 16 + ...
    #pragma unroll
    for (int ci = 0; ci < 4; ++ci) {
        const int n = colBase + ci * 16 + lo;
        const float bval = HAS_BIAS ? bias[n] : 0.0f;
        #pragma unroll
        for (int vg = 0; vg < 8; ++vg) {
            const int r0 = rowTile * 32 + vg + 8 * hi;
            float v0 = accA[ci][vg] + bval;
            float v1 = accB[ci][vg] + bval;
            if (ACT_TANH) { v0 = tanhf(v0); v1 = tanhf(v1); }
            C[(long)r0 * ldc + n] = v0;
            C[(long)(r0 + 16) * ldc + n] = v1;
        }
    }
}

// ---------------------------------------------------------------------------
// Scores: one block per (t,b).  score[s] = sum_h tanh(qp[h] + ap[s][h]) * v[h]
// then masked softmax over s.  blockDim.x == S_DIM == 128 (thread == s).
// Annotation-projection rows are staged through LDS in 64-wide h-chunks with
// coalesced global reads; +1 padding keeps the compute phase bank-conflict-free.
// ---------------------------------------------------------------------------
__global__ __launch_bounds__(128) void mlp_attn_scores(
    const float* __restrict__ qp,    // (T,B,H) q-projection
    const float* __restrict__ ap,    // (S,B,H) annotation-projection
    const float* __restrict__ v,     // (H)
    const int*   __restrict__ lens,  // (B)
    float* __restrict__ scores)      // (T,B,S)
{
    __shared__ float sh_q[H_DIM];
    __shared__ float sh_v[H_DIM];
    __shared__ float sh_a[S_DIM * (HCHUNK + 1)];   // 128 x 65 floats
    __shared__ float red[S_DIM];

    const int tid = threadIdx.x;
    const int b = blockIdx.x % B_DIM;
    const int t = blockIdx.x / B_DIM;

    const float* qrow = qp + ((long)t * B_DIM + b) * H_DIM;
    for (int i = tid; i < H_DIM; i += 128) {
        sh_q[i] = qrow[i];
        sh_v[i] = v[i];
    }

    const int s = tid;
    float acc = 0.0f;

    for (int h0 = 0; h0 < H_DIM; h0 += HCHUNK) {
        __syncthreads();
        // stage chunk: 128 rows x 64 floats, 128B-contiguous per wave
        for (int e = tid; e < S_DIM * HCHUNK; e += 128) {
            const int r  = e >> 6;     // row (s index)
            const int hh = e & 63;
            sh_a[r * (HCHUNK + 1) + hh] = ap[((long)r * B_DIM + b) * H_DIM + h0 + hh];
        }
        __syncthreads();
        const float* arow = &sh_a[s * (HCHUNK + 1)];
        #pragma unroll 4
        for (int hh = 0; hh < HCHUNK; ++hh)
            acc += tanhf(sh_q[h0 + hh] + arow[hh]) * sh_v[h0 + hh];
    }

    const int len = lens[b];
    const bool valid = (s < len);
    float sc = valid ? acc : -INFINITY;

    // max-reduce over the 128 scores
    __syncthreads();
    red[tid] = sc;
    __syncthreads();
    for (int off = 64; off > 0; off >>= 1) {
        if (tid < off) red[tid] = fmaxf(red[tid], red[tid + off]);
        __syncthreads();
    }
    const float mx = red[0];
    __syncthreads();

    const float e = valid ? __expf(sc - mx) : 0.0f;
    red[tid] = e;
    __syncthreads();
    for (int off = 64; off > 0; off >>= 1) {
        if (tid < off) red[tid] += red[tid + off];
        __syncthreads();
    }
    const float denom = red[0];

    scores[((long)t * B_DIM + b) * S_DIM + s] = e / denom;
}

// ---------------------------------------------------------------------------
extern "C" void kernel_launch(void* const* d_in, const int* in_sizes, int n_in,
                              void* d_out, int out_size, void* d_ws, size_t ws_size,
                              hipStream_t stream)
{
    const float* queries     = (const float*)d_in[0];  // (T,B,H)
    const float* annotations = (const float*)d_in[1];  // (S,B,A)
    const int*   lens        = (const int*)  d_in[2];  // (B)
    const float* Wq          = (const float*)d_in[3];  // (H,H)
    const float* bq          = (const float*)d_in[4];  // (H)
    const float* Wa          = (const float*)d_in[5];  // (H,A)
    const float* v           = (const float*)d_in[6];  // (H)
    const float* Wattn       = (const float*)d_in[7];  // (H, H+A)
    const float* battn       = (const float*)d_in[8];  // (H)

    float* states = (float*)d_out;                         // (T,B,H)
    float* scores = states + (long)T_DIM * B_DIM * H_DIM;  // (T,B,S)

    float* qp  = (float*)d_ws;                             // (T*B, H) = 2 MB
    float* apj = qp  + (long)T_DIM * B_DIM * H_DIM;        // (S*B, H) = 2 MB
    float* ctx = apj + (long)S_DIM * B_DIM * H_DIM;        // (T*B, A) = 2 MB

    const dim3 blk(128);
    const int M_qa = T_DIM * B_DIM;  // 1024 rows for the big GEMMs

    // 1) q_proj = queries @ Wq^T + bq        M=1024 N=512 K=512 -> 256 waves
    {
        int nblk = ((M_qa / 32) * (H_DIM / 64)) / WAVES_PER_BLOCK;  // 64
        wmma_gemm_f32<false, false, false, true><<<dim3(nblk, 1), blk, 0, stream>>>(
            queries, H_DIM, nullptr, 0, H_DIM, Wq, H_DIM, bq,
            qp, H_DIM, M_qa, H_DIM, H_DIM, 0, 0, 0);
    }
    // 2) a_proj = annotations @ Wa^T          M=1024 N=512 K=512
    {
        int nblk = ((M_qa / 32) * (H_DIM / 64)) / WAVES_PER_BLOCK;  // 64
        wmma_gemm_f32<false, false, false, false><<<dim3(nblk, 1), blk, 0, stream>>>(
            annotations, A_DIM, nullptr, 0, A_DIM, Wa, A_DIM, nullptr,
            apj, H_DIM, M_qa, H_DIM, A_DIM, 0, 0, 0);
    }
    // 3) scores = softmax_s( v . tanh(q_proj + a_proj) ), masked by lens
    mlp_attn_scores<<<dim3(T_DIM * B_DIM), blk, 0, stream>>>(qp, apj, v, lens, scores);

    // 4) contexts[b] = scores_b (T x S) @ annotations_b (S x A), batched over b
    //    Xa = scores + b*S, lda = B*S ; W = annotations + b*A (row-major), ldw = B*A
    //    C  = ctx + b*A -> ctx[(t*B+b)*A + n]
    {
        int nblk = ((T_DIM / 32) * (A_DIM / 64)) / WAVES_PER_BLOCK;  // 8
        wmma_gemm_f32<true, false, false, false><<<dim3(nblk, B_DIM), blk, 0, stream>>>(
            scores, B_DIM * S_DIM, nullptr, 0, S_DIM,
            annotations, B_DIM * A_DIM, nullptr,
            ctx, B_DIM * A_DIM, T_DIM, A_DIM, S_DIM,
            (long)S_DIM, (long)A_DIM, (long)A_DIM);
    }
    // 5) states = tanh( [queries | ctx] @ Wattn^T + battn )   M=1024 N=512 K=1024
    {
        int nblk = ((M_qa / 32) * (H_DIM / 64)) / WAVES_PER_BLOCK;  // 64
        wmma_gemm_f32<false, true, true, true><<<dim3(nblk, 1), blk, 0, stream>>>(
            queries, H_DIM, ctx, A_DIM, H_DIM, Wattn, H_DIM + A_DIM, battn,
            states, H_DIM, M_qa, H_DIM, H_DIM + A_DIM, 0, 0, 0);
    }
}